// PrototypeClassifier_63367947485311
// MI455X (gfx1250) — compile-verified
//
#include <hip/hip_runtime.h>
#include <hip/hip_bf16.h>

#define LATENT 40
#define NB 32768
#define NP 2048
#define NO 200

typedef float v2f __attribute__((ext_vector_type(2)));
typedef float v8f __attribute__((ext_vector_type(8)));

// ---------- workspace layout (in floats) ----------
// [0..199]     wsum
// [256..295]   proto_sum  (16B aligned: 256*4 = 1024)
// [304]        proto_sq
// [305]        min bits (as uint)
// [320..447]   per-block partial sums (128 blocks)
// [512..]      dvec (32768)
#define WS_WSUM   0
#define WS_PSUM   256
#define WS_PSQ    304
#define WS_MIN    305
#define WS_BSUM   320
#define WS_DVEC   512

__global__ void pc_init_kernel(unsigned* minbits) {
    *minbits = 0x7F800000u; // +inf
}

// wsum[o] = sum_p w[o,p]   (one block per output row, coalesced reads)
__global__ void pc_wsum_kernel(const float* __restrict__ w, float* __restrict__ wsum) {
    __shared__ float s[256];
    const int o = blockIdx.x;
    const float* row = w + (size_t)o * NP;
    float acc = 0.0f;
    for (int k = threadIdx.x; k < NP; k += 256) acc += row[k];
    s[threadIdx.x] = acc;
    __syncthreads();
    for (int off = 128; off > 0; off >>= 1) {
        if (threadIdx.x < off) s[threadIdx.x] += s[threadIdx.x + off];
        __syncthreads();
    }
    if (threadIdx.x == 0) wsum[o] = s[0];
}

// proto_sum[d] = column sums of prototypes; proto_sq = Frobenius sum of squares
__global__ void pc_proto_kernel(const float* __restrict__ protos,
                                float* __restrict__ proto_sum,
                                float* __restrict__ proto_sq) {
    __shared__ float ssq[64];
    const int t = threadIdx.x;
    float sum = 0.0f, sq = 0.0f;
    if (t < LATENT) {
        for (int r = 0; r < NP; ++r) {
            float v = protos[r * LATENT + t];
            sum += v;
            sq += v * v;
        }
        proto_sum[t] = sum;
    }
    ssq[t] = (t < LATENT) ? sq : 0.0f;
    __syncthreads();
    for (int off = 32; off > 0; off >>= 1) {
        if (t < off) ssq[t] += ssq[t + off];
        __syncthreads();
    }
    if (t == 0) *proto_sq = ssq[0];
}

// d[i] = sqrt(P*||x_i||^2 - 2*x_i.proto_sum + proto_sq); block min + block sum
__global__ void pc_drow_kernel(const float* __restrict__ input,
                               const float* __restrict__ proto_sum,
                               const float* __restrict__ proto_sq_p,
                               float* __restrict__ dvec,
                               float* __restrict__ blocksums,
                               unsigned* __restrict__ minbits) {
    const int i = blockIdx.x * 256 + threadIdx.x;
    const float4* row = (const float4*)(input + (size_t)i * LATENT);
    const float4* ps  = (const float4*)proto_sum;
    float q = 0.0f, dot = 0.0f;
#pragma unroll
    for (int k = 0; k < LATENT / 4; ++k) {
        float4 x = row[k];
        float4 p = ps[k];
        q   += x.x * x.x + x.y * x.y + x.z * x.z + x.w * x.w;
        dot += x.x * p.x + x.y * p.y + x.z * p.z + x.w * p.w;
    }
    float s = (float)NP * q - 2.0f * dot + *proto_sq_p;
    float d = sqrtf(s);
    dvec[i] = d;

    __shared__ float smin[256];
    __shared__ float ssum[256];
    smin[threadIdx.x] = d;
    ssum[threadIdx.x] = d;
    __syncthreads();
    for (int off = 128; off > 0; off >>= 1) {
        if (threadIdx.x < off) {
            smin[threadIdx.x] = fminf(smin[threadIdx.x], smin[threadIdx.x + off]);
            ssum[threadIdx.x] += ssum[threadIdx.x + off];
        }
        __syncthreads();
    }
    if (threadIdx.x == 0) {
        blocksums[blockIdx.x] = ssum[0];
        // d >= 0 so float bit pattern is monotone as unsigned
        atomicMin(minbits, __float_as_uint(smin[0]));
    }
}

__global__ void pc_finalize_kernel(const unsigned* __restrict__ minbits,
                                   const float* __restrict__ blocksums,
                                   float* __restrict__ out01) {
    float s = 0.0f;
    for (int k = 0; k < NB / 256; ++k) s += blocksums[k];  // deterministic order
    out01[0] = __uint_as_float(*minbits);                  // min1 = min_i d[i]
    out01[1] = s * (1.0f / (float)NB);                     // min2 = mean_i d[i]
}

// out[i,o] = d[i]*wsum[o] + b[o] via V_WMMA_F32_16X16X4_F32 rank-1 tiles.
// A: every K slot = d[row]  (16x4, all columns equal -> layout-robust)
// B: every slot   = wsum[col]/4 (sum over K=4 restores d*wsum)
// C: b[col] broadcast into all 8 accumulator VGPRs
__global__ void pc_outer_kernel(const float* __restrict__ dvec,
                                const float* __restrict__ wsum,
                                const float* __restrict__ bvec,
                                float* __restrict__ out) {
    const int lane = threadIdx.x & 31;
    const int wave = threadIdx.x >> 5;
    const int l15  = lane & 15;
    const int row_base = blockIdx.x * 16;

    const float dval = dvec[row_base + l15];
    v2f a;
    a[0] = dval;
    a[1] = dval;

    const int rstore = row_base + ((lane >> 4) << 3); // +8 for lanes 16..31

    for (int ct = wave; ct < (NO + 15) / 16; ct += 8) {
        const int n  = ct * 16 + l15;
        const int nc = (n < NO) ? n : (NO - 1);
        const float wv = 0.25f * wsum[nc];
        const float bv = bvec[nc];

        v2f bfrag;
        bfrag[0] = wv;
        bfrag[1] = wv;
        v8f c;
#pragma unroll
        for (int j = 0; j < 8; ++j) c[j] = bv;

        // EXEC is all-1s here (divergence only happens after the WMMA)
        v8f dacc = __builtin_amdgcn_wmma_f32_16x16x4_f32(
            false, a, false, bfrag, (short)0, c, false, false);

        if (n < NO) {
#pragma unroll
            for (int j = 0; j < 8; ++j) {
                // C/D layout: VGPR j -> row j (lanes 0-15) / row j+8 (lanes 16-31)
                out[(size_t)(rstore + j) * NO + n] = dacc[j];
            }
        }
    }
}

extern "C" void kernel_launch(void* const* d_in, const int* in_sizes, int n_in,
                              void* d_out, int out_size, void* d_ws, size_t ws_size,
                              hipStream_t stream) {
    const float* input  = (const float*)d_in[0]; // (32768, 40)
    const float* protos = (const float*)d_in[1]; // (2048, 40)
    const float* w      = (const float*)d_in[2]; // (200, 2048)
    const float* b      = (const float*)d_in[3]; // (200,)
    float* outf = (float*)d_out;                 // [min1, min2, out(32768x200)]

    float* ws        = (float*)d_ws;
    float* wsum      = ws + WS_WSUM;
    float* proto_sum = ws + WS_PSUM;
    float* proto_sq  = ws + WS_PSQ;
    unsigned* minbits = (unsigned*)(ws + WS_MIN);
    float* blocksums = ws + WS_BSUM;
    float* dvec      = ws + WS_DVEC;

    pc_init_kernel<<<1, 1, 0, stream>>>(minbits);
    pc_wsum_kernel<<<NO, 256, 0, stream>>>(w, wsum);
    pc_proto_kernel<<<1, 64, 0, stream>>>(protos, proto_sum, proto_sq);
    pc_drow_kernel<<<NB / 256, 256, 0, stream>>>(input, proto_sum, proto_sq,
                                                 dvec, blocksums, minbits);
    pc_finalize_kernel<<<1, 1, 0, stream>>>(minbits, blocksums, outf);
    pc_outer_kernel<<<NB / 16, 256, 0, stream>>>(dvec, wsum, b, outf + 2);
}